// DNM_Linear_55740085567636
// MI455X (gfx1250) — compile-verified
//
#include <hip/hip_runtime.h>

// DNM_Linear for MI455X (gfx1250), compile-only tuned. Round 2.
//
// Shapes (reference): BATCH=128, OUT=256, M=8, IN=512.
// z[i,b] = prod_j sum_k sigmoid(x[i,j,k]*W[b,j,k]); per-row sum-normalize,
// then standardize with unbiased (ddof=1) std.
//
// No bilinear structure (sigmoid inside K-reduction) => WMMA inapplicable.
// Transcendental-bound: ~268M v_exp/v_rcp. CDNA5 paths used:
//   - global_load_async_to_lds_b128 + s_wait_asynccnt (x-row staging)
//   - ds_load_b128 LDS reads, wave32 __shfl_xor butterflies
//   - v_exp_f32 / v_rcp_f32 / v_rsq_f32 fast transcendentals
// Round-2 changes: 4 independent sigmoid accumulator chains per j (trans ILP),
// shuffle-based block reductions (2 barriers each instead of 8).

#define BATCH 128
#define OUTN  256
#define MBR   8
#define INSZ  512
#define ROWE  (MBR * INSZ)        // 4096 floats = 16 KB per row
#define NWAVE (OUTN / 32)         // 8 waves per block

__device__ __forceinline__ float fast_sigmoid(float t) {
    // sigmoid(t) = 1 / (1 + exp2(-t*log2(e)))  -> v_exp_f32 + v_add + v_rcp_f32
    float e = __builtin_amdgcn_exp2f(-1.44269504088896340736f * t);
    return __builtin_amdgcn_rcpf(1.0f + e);
}

__device__ __forceinline__ float wave_reduce_add(float v) {
    // wave32 butterfly: 5 cross-lane steps, no barriers
    #pragma unroll
    for (int m = 16; m > 0; m >>= 1) v += __shfl_xor(v, m, 32);
    return v;
}

__global__ __launch_bounds__(OUTN) void dnm_linear_kernel(
    const float* __restrict__ x,   // [BATCH, M, IN]
    const float* __restrict__ w,   // [OUT,   M, IN]
    float* __restrict__ out)       // [BATCH, OUT]
{
    __shared__ __align__(16) float xs[ROWE];   // 16 KB x[i,:,:]; scratch later

    const int i    = blockIdx.x;       // batch row
    const int t    = threadIdx.x;      // output column b
    const int lane = t & 31;
    const int wid  = t >> 5;

    // ---- Stage x[i,:,:] into LDS with CDNA5 async copies (ASYNCcnt path) ----
    {
        const float* gsrc = x + (size_t)i * ROWE;
        #pragma unroll
        for (int it = 0; it < (ROWE / 4) / OUTN; ++it) {    // 4 b128 per thread
            const int slot = t + it * OUTN;                 // float4 slot index
            // Low 32 bits of the flat address of an LDS object are the
            // wave-relative LDS byte offset (ISA 10.2 aperture mapping).
            unsigned lds_off = (unsigned)(size_t)(&xs[slot * 4]);
            const float* gp = gsrc + slot * 4;
            asm volatile("global_load_async_to_lds_b128 %0, %1, off"
                         :: "v"(lds_off), "v"(gp)
                         : "memory");
        }
        asm volatile("s_wait_asynccnt 0" ::: "memory");
    }
    __syncthreads();

    // ---- Per-thread: z = prod_j sum_k sigmoid(x*w), W streamed as b128 ----
    const float4* __restrict__ wrow = (const float4*)(w + (size_t)t * ROWE);
    const float4* __restrict__ xv4  = (const float4*)xs;

    float prod = 1.0f;
    #pragma unroll
    for (int j = 0; j < MBR; ++j) {
        // 4 independent accumulators -> 4 parallel trans dependency chains
        float s0 = 0.0f, s1 = 0.0f, s2 = 0.0f, s3 = 0.0f;
        #pragma unroll 4
        for (int k = 0; k < INSZ / 4; ++k) {
            const int v = j * (INSZ / 4) + k;
            float4 wv = wrow[v];     // global_load_b128 (L2 resident, 8x line reuse)
            float4 xv = xv4[v];      // ds_load_b128
            s0 += fast_sigmoid(xv.x * wv.x);
            s1 += fast_sigmoid(xv.y * wv.y);
            s2 += fast_sigmoid(xv.z * wv.z);
            s3 += fast_sigmoid(xv.w * wv.w);
        }
        prod *= (s0 + s1) + (s2 + s3);
    }

    // ---- Fused row normalization over b (whole row lives in this block) ----
    __syncthreads();                 // all xs reads done; reuse LDS as scratch
    float* red = xs;

    // block_sum: wave butterfly + 8-entry LDS combine (2 barriers total)
    auto block_sum = [&](float v) -> float {
        v = wave_reduce_add(v);
        if (lane == 0) red[wid] = v;
        __syncthreads();
        float s = 0.0f;
        #pragma unroll
        for (int u = 0; u < NWAVE; ++u) s += red[u];
        __syncthreads();             // red free for next round
        return s;
    };

    const float total = block_sum(prod);
    const float zn    = prod * __builtin_amdgcn_rcpf(total);

    const float mean  = block_sum(zn) * (1.0f / OUTN);
    const float d     = zn - mean;

    const float var     = block_sum(d * d) * (1.0f / (OUTN - 1));
    const float inv_std = __builtin_amdgcn_rsqf(var);   // v_rsq_f32

    out[(size_t)i * OUTN + t] = d * inv_std;
}

extern "C" void kernel_launch(void* const* d_in, const int* in_sizes, int n_in,
                              void* d_out, int out_size, void* d_ws, size_t ws_size,
                              hipStream_t stream) {
    (void)in_sizes; (void)n_in; (void)d_ws; (void)ws_size; (void)out_size;
    const float* x = (const float*)d_in[0];   // [128, 8, 512] f32
    const float* w = (const float*)d_in[1];   // [256, 8, 512] f32
    float* out = (float*)d_out;               // [128, 256]    f32

    dnm_linear_kernel<<<dim3(BATCH), dim3(OUTN), 0, stream>>>(x, w, out);
}